// BoxEstimationNet_20633022890028
// MI455X (gfx1250) — compile-verified
//
#include <hip/hip_runtime.h>
#include <hip/hip_bf16.h>

// ---------------------------------------------------------------------------
// BoxEstimationNet on MI455X (gfx1250), wave32, f32 WMMA (16x16x4).
// ~63 GFLOP, ~20MB working set -> L2 resident. Each SA stage (ball query +
// 3 MLP layers + max-pool) is one fused kernel with LDS-resident activations.
// Weights pre-packed as paired WMMA B-fragments: one global_load_b128 feeds
// two v_wmma_f32_16x16x4_f32; K depth is a template constant so the inner
// loop unrolls into load-clause + WMMA bursts.
// ---------------------------------------------------------------------------

typedef float v2f __attribute__((ext_vector_type(2)));
typedef float v4f __attribute__((ext_vector_type(4)));
typedef float v8f __attribute__((ext_vector_type(8)));

#define BATCH   128
#define NPOINTS 2048

// layer table: 0-2 sa1, 3-5 sa2, 6-8 sa3, 9-11 fc
__constant__ int d_CIN[12]   = {3,64,64, 131,128,128, 259,256,256, 515,512,256};
__constant__ int d_COUT[12]  = {64,64,128, 128,128,256, 256,256,512, 512,256,59};
__constant__ int d_CHOFF[12] = {0,64,128, 256,384,512, 768,1024,1280, 1792,2304,2560};
__constant__ int d_KCH[12]   = {1,16,16, 33,32,32, 65,64,64, 129,128,64};
__constant__ int d_NT[12]    = {4,4,8, 8,8,16, 16,16,32, 32,16,4};
// packed offsets: size per layer = NT * kchp * 64 floats (kchp = kch rounded even)
__constant__ int d_POFF[12]  = {0,512,4608, 12800,30208,46592, 79360,146944,212480,
                                343552,609792,740864};
static const int h_POFF[12]  = {0,512,4608, 12800,30208,46592, 79360,146944,212480,
                                343552,609792,740864};
static constexpr size_t WPACK_TOTAL = 757248 + 64;   // floats (+tail pad)

static constexpr size_t CTOT     = 2619;
static constexpr size_t OFF_S    = 0;
static constexpr size_t OFF_T    = CTOT;
static constexpr size_t OFF_XYZ1 = 2 * CTOT;
static constexpr size_t OFF_XYZ2 = OFF_XYZ1 + (size_t)BATCH * 128 * 3;
static constexpr size_t OFF_F1   = OFF_XYZ2 + (size_t)BATCH * 32 * 3;
static constexpr size_t OFF_F2   = OFF_F1   + (size_t)BATCH * 128 * 128;
static constexpr size_t OFF_X1   = OFF_F2   + (size_t)BATCH * 32 * 256;   // stride 516
static constexpr size_t OFF_H1   = OFF_X1   + (size_t)BATCH * 516;
static constexpr size_t OFF_H2   = OFF_H1   + (size_t)BATCH * 512;
static constexpr size_t OFF_WP   = OFF_H2   + (size_t)BATCH * 256;

struct BNPtrs {
    const float* b[12];
    const float* beta[12];
    const float* gamma[12];
    const float* mean[12];
    const float* var[12];
};
struct WPtrs { const float* W[12]; };

// --------------------------- BN fusion: s,t per channel ---------------------
__global__ void fuse_bn_kernel(BNPtrs P, float* __restrict__ S, float* __restrict__ T) {
    int ly = blockIdx.x;
    int c  = threadIdx.x;
    if (c >= d_COUT[ly]) return;
    int off = d_CHOFF[ly];
    if (ly == 11) {                       // last FC layer: no BN, no relu
        S[off + c] = 1.0f;
        T[off + c] = P.b[11][c];
        return;
    }
    float inv = rsqrtf(P.var[ly][c] + 1e-5f);
    float sc  = P.gamma[ly][c] * inv;
    S[off + c] = sc;
    T[off + c] = (P.b[ly][c] - P.mean[ly][c]) * sc + P.beta[ly][c];
}

// ----------------- pack weights into paired WMMA B-fragments ----------------
// layout: [nt][kpair][lane][4]; q0,q1 = chunk 2*kp (K halves), q2,q3 = chunk 2*kp+1
// lane: n = nt*16+(lane&15), kk = chunk*4+((lane>>4)<<1)+half
__global__ void pack_w_kernel(WPtrs P, float* __restrict__ Wp) {
    const int ly   = blockIdx.x;
    const float* W = P.W[ly];
    const int Cin = d_CIN[ly], Cout = d_COUT[ly];
    const int kch = d_KCH[ly], nt_n = d_NT[ly];
    const int kchp = (kch + 1) & ~1;
    float* dst = Wp + d_POFF[ly];
    const int total = nt_n * (kchp >> 1) * 128;
    for (int e = threadIdx.x; e < total; e += blockDim.x) {
        int q    = e & 3;
        int lane = (e >> 2) & 31;
        int blk  = e >> 7;              // nt*(kchp/2) + kp
        int kp = blk % (kchp >> 1), nt = blk / (kchp >> 1);
        int chunk = 2 * kp + (q >> 1);
        int half  = q & 1;
        int n  = nt * 16 + (lane & 15);
        int kk = chunk * 4 + ((lane >> 4) << 1) + half;
        dst[e] = (kk < Cin && n < Cout) ? W[(size_t)kk * Cout + n] : 0.f;
    }
}

// --------------------------- farthest point sampling ------------------------
template <int NP, int NS, int NT>
__global__ void fps_kernel(const float* __restrict__ xyz, float* __restrict__ out) {
    __shared__ float dist[NP];
    __shared__ float rv[NT];
    __shared__ int   ri[NT];
    __shared__ int   s_last;
    const int b   = blockIdx.x;
    const int tid = threadIdx.x;
    const float* P = xyz + (size_t)b * NP * 3;
    float* O = out + (size_t)b * NS * 3;

    for (int i = tid; i < NP; i += NT) dist[i] = 1e10f;
    if (tid == 0) { s_last = 0; O[0] = P[0]; O[1] = P[1]; O[2] = P[2]; }
    __syncthreads();

    for (int it = 1; it < NS; ++it) {
        int last = s_last;
        float px = P[last*3+0], py = P[last*3+1], pz = P[last*3+2];
        float bv = -1.0f; int bi = 0;
        for (int i = tid; i < NP; i += NT) {
            float dx = P[i*3+0]-px, dy = P[i*3+1]-py, dz = P[i*3+2]-pz;
            float d  = dx*dx + dy*dy + dz*dz;
            float dm = fminf(dist[i], d);
            dist[i]  = dm;
            if (dm > bv) { bv = dm; bi = i; }
        }
        rv[tid] = bv; ri[tid] = bi;
        __syncthreads();
        for (int s = NT/2; s > 0; s >>= 1) {
            if (tid < s) {
                if (rv[tid+s] > rv[tid] ||
                    (rv[tid+s] == rv[tid] && ri[tid+s] < ri[tid])) {
                    rv[tid] = rv[tid+s]; ri[tid] = ri[tid+s];
                }
            }
            __syncthreads();
        }
        if (tid == 0) {
            int n = ri[0];
            s_last = n;
            O[it*3+0] = P[n*3+0]; O[it*3+1] = P[n*3+1]; O[it*3+2] = P[n*3+2];
        }
        __syncthreads();
    }
}

// --------------------------- WMMA tiled MLP layer ---------------------------
// A: (M x >=4*KCH) in LDS, row stride lda, zero-padded K tail.
// Wp: packed paired fragments for this layer. Cout must be multiple of 16.
// D = relu((A@W)*s + t).  FUSEMAX: fold max over M rows -> atomicMax Out[col].
template <bool FUSEMAX, int KCH>
__device__ inline void mlp_tiles(const float* __restrict__ A, int lda, int M,
                                 const float* __restrict__ Wp, int Cout,
                                 const float* __restrict__ sC, const float* __restrict__ tC,
                                 float* __restrict__ Out, int ldo,
                                 int wave, int lane, int nWaves) {
    constexpr int KPAIRS = KCH >> 1;           // full pairs
    constexpr int KPT    = (KCH + 1) >> 1;     // pairs incl. zero-padded tail
    const int mt_n = M >> 4;
    const int nt_n = Cout >> 4;
    const int mrow  = lane & 15;
    const int khalf = (lane >> 4) << 1;

    for (int tile = wave; tile < mt_n * nt_n; tile += nWaves) {
        const int m0 = (tile % mt_n) << 4;
        const int nt = tile / mt_n;
        const float* Arow = A + (size_t)(m0 + mrow) * lda + khalf;
        const v4f*   Bq   = (const v4f*)Wp + (size_t)nt * KPT * 32 + lane;
        v8f acc0 = {0.f,0.f,0.f,0.f,0.f,0.f,0.f,0.f};
        v8f acc1 = {0.f,0.f,0.f,0.f,0.f,0.f,0.f,0.f};
        if constexpr (KPAIRS > 0) {
#pragma unroll 8
            for (int kp = 0; kp < KPAIRS; ++kp) {
                v4f bb = Bq[(size_t)kp * 32];            // 16B coalesced
                const float* Ac = Arow + kp * 8;
                v2f a0; a0.x = Ac[0]; a0.y = Ac[1];
                v2f a1; a1.x = Ac[4]; a1.y = Ac[5];
                v2f b0; b0.x = bb.x;  b0.y = bb.y;
                v2f b1; b1.x = bb.z;  b1.y = bb.w;
                acc0 = __builtin_amdgcn_wmma_f32_16x16x4_f32(
                           false, a0, false, b0, (short)0, acc0, false, false);
                acc1 = __builtin_amdgcn_wmma_f32_16x16x4_f32(
                           false, a1, false, b1, (short)0, acc1, false, false);
            }
        }
        if constexpr (KCH & 1) {                         // zero-padded tail pair
            v4f bb = Bq[(size_t)KPAIRS * 32];
            const float* Ac = Arow + KPAIRS * 8;
            v2f a; a.x = Ac[0]; a.y = Ac[1];
            v2f b0; b0.x = bb.x; b0.y = bb.y;
            acc0 = __builtin_amdgcn_wmma_f32_16x16x4_f32(
                       false, a, false, b0, (short)0, acc0, false, false);
        }
        v8f acc;
        if constexpr (KPAIRS > 0) acc = acc0 + acc1; else acc = acc0;

        const int   n  = (nt << 4) + mrow;
        const float sc = sC[n];
        const float tc = tC[n];
        const int   mh = (lane >> 4) << 3;   // lanes 16-31 hold rows M=8..15
        if (FUSEMAX) {
            float mx = 0.f;
#pragma unroll
            for (int r = 0; r < 8; ++r) {
                float v = fmaxf(acc[r] * sc + tc, 0.f);  // relu always here
                mx = fmaxf(mx, v);
            }
            mx = fmaxf(mx, __shfl_xor(mx, 16, 32));      // merge row halves
            if (lane < 16)
                atomicMax((int*)&Out[n], __float_as_int(mx));  // vals >= 0
        } else {
#pragma unroll
            for (int r = 0; r < 8; ++r) {
                float v = fmaxf(acc[r] * sc + tc, 0.f);  // relu (all hidden layers)
                Out[(size_t)(m0 + mh + r) * ldo + n] = v;
            }
        }
    }
}

// --------------------------- SA stage 1 -------------------------------------
// block = (b, s in 0..127), 128 threads (4 waves). 2048 pts -> 64 nbrs -> MLP
// 3->64->64->128 -> max  => f1[b,s,0:128]
__global__ void sa1_kernel(const float* __restrict__ pts, const float* __restrict__ xyz1,
                           const float* __restrict__ Wp,
                           const float* __restrict__ S, const float* __restrict__ T,
                           float* __restrict__ f1) {
    extern __shared__ float smem[];
    float* X  = smem;                 // 64 x 9  (cols 0-2 data, 3-7 zero: kchp=2)
    float* B1 = X  + 64 * 9;          // 64 x 65
    float* B2 = B1 + 64 * 65;         // 64 x 65
    float* CM = B2 + 64 * 65;         // 128
    int*  NBR = (int*)(CM + 128);     // 64
    __shared__ int s_cnt;

    const int tid = threadIdx.x, lane = tid & 31, wave = tid >> 5;
    const int b = blockIdx.x >> 7;
    const int s = blockIdx.x & 127;
    const float* P = pts + (size_t)b * NPOINTS * 3;
    const float cx = xyz1[((size_t)b*128 + s)*3 + 0];
    const float cy = xyz1[((size_t)b*128 + s)*3 + 1];
    const float cz = xyz1[((size_t)b*128 + s)*3 + 2];

    CM[tid] = 0.f;                    // 128 entries, 128 threads

    if (wave == 0) {                  // ordered ball query, wave32 compaction
        int count = 0;
        for (int base = 0; base < NPOINTS && count < 64; base += 32) {
            int i = base + lane;
            float dx = P[i*3+0]-cx, dy = P[i*3+1]-cy, dz = P[i*3+2]-cz;
            bool in = (dx*dx + dy*dy + dz*dz) <= 0.04f;
            unsigned m = (unsigned)__ballot(in);
            int pos = count + __popc(m & ((1u << lane) - 1u));
            if (in && pos < 64) NBR[pos] = i;
            count += __popc(m);
        }
        if (lane == 0) s_cnt = count;
    }
    __syncthreads();
    int cnt = s_cnt; if (cnt > 64) cnt = 64;
    int first = (cnt > 0) ? NBR[0] : 0;
    for (int p = cnt + tid; p < 64; p += 128) NBR[p] = first;
    __syncthreads();

    if (tid < 64) {                   // gather relative xyz, zero-pad K
        int pi = NBR[tid];
        X[tid*9+0] = P[pi*3+0]-cx;
        X[tid*9+1] = P[pi*3+1]-cy;
        X[tid*9+2] = P[pi*3+2]-cz;
        X[tid*9+3] = 0.f; X[tid*9+4] = 0.f; X[tid*9+5] = 0.f;
        X[tid*9+6] = 0.f; X[tid*9+7] = 0.f;
    }
    __syncthreads();

    mlp_tiles<false, 1 >(X,  9, 64, Wp + 0,     64, S+0,   T+0,   B1, 65, wave, lane, 4);
    __syncthreads();
    mlp_tiles<false, 16>(B1, 65, 64, Wp + 512,  64, S+64,  T+64,  B2, 65, wave, lane, 4);
    __syncthreads();
    mlp_tiles<true,  16>(B2, 65, 64, Wp + 4608, 128, S+128, T+128, CM,  0, wave, lane, 4);
    __syncthreads();

    f1[((size_t)b*128 + s)*128 + tid] = CM[tid];
}

// --------------------------- SA stage 2 -------------------------------------
// block = (b, s in 0..31). 128 xyz1 pts -> 64 nbrs, feats 3+128 -> MLP
// 131->128->128->256 -> max => f2[b,s,0:256]
__global__ void sa2_kernel(const float* __restrict__ xyz1, const float* __restrict__ f1,
                           const float* __restrict__ xyz2,
                           const float* __restrict__ Wp,
                           const float* __restrict__ S, const float* __restrict__ T,
                           float* __restrict__ f2) {
    extern __shared__ float smem[];
    float* X  = smem;                  // 64 x 137 (0-2 xyz, 3-130 feats, 131-135 zero)
    float* B1 = X  + 64 * 137;         // 64 x 129
    float* B2 = B1 + 64 * 129;         // 64 x 129
    float* CM = B2 + 64 * 129;         // 256
    int*  NBR = (int*)(CM + 256);      // 64
    __shared__ int s_cnt;

    const int tid = threadIdx.x, lane = tid & 31, wave = tid >> 5;
    const int b = blockIdx.x >> 5;
    const int s = blockIdx.x & 31;
    const float* P1  = xyz1 + (size_t)b * 128 * 3;
    const float* F1b = f1   + (size_t)b * 128 * 128;
    const float cx = xyz2[((size_t)b*32 + s)*3 + 0];
    const float cy = xyz2[((size_t)b*32 + s)*3 + 1];
    const float cz = xyz2[((size_t)b*32 + s)*3 + 2];

    for (int i = tid; i < 256; i += 128) CM[i] = 0.f;

    if (wave == 0) {
        int count = 0;
        for (int base = 0; base < 128 && count < 64; base += 32) {
            int i = base + lane;
            float dx = P1[i*3+0]-cx, dy = P1[i*3+1]-cy, dz = P1[i*3+2]-cz;
            bool in = (dx*dx + dy*dy + dz*dz) <= 0.16f;
            unsigned m = (unsigned)__ballot(in);
            int pos = count + __popc(m & ((1u << lane) - 1u));
            if (in && pos < 64) NBR[pos] = i;
            count += __popc(m);
        }
        if (lane == 0) s_cnt = count;
    }
    __syncthreads();
    int cnt = s_cnt; if (cnt > 64) cnt = 64;
    int first = (cnt > 0) ? NBR[0] : 0;
    for (int p = cnt + tid; p < 64; p += 128) NBR[p] = first;
    __syncthreads();

    for (int e = tid; e < 64 * 136; e += 128) {
        int row = e / 136, col = e - row * 136;
        int pi = NBR[row];
        float v;
        if (col < 3)        v = P1[pi*3 + col] - ((col==0)?cx:((col==1)?cy:cz));
        else if (col < 131) v = F1b[(size_t)pi*128 + (col - 3)];
        else                v = 0.f;
        X[(size_t)row*137 + col] = v;
    }
    __syncthreads();

    mlp_tiles<false, 33>(X, 137, 64, Wp + 12800, 128, S+256, T+256, B1, 129, wave, lane, 4);
    __syncthreads();
    mlp_tiles<false, 32>(B1, 129, 64, Wp + 30208, 128, S+384, T+384, B2, 129, wave, lane, 4);
    __syncthreads();
    mlp_tiles<true,  32>(B2, 129, 64, Wp + 46592, 256, S+512, T+512, CM, 0,   wave, lane, 4);
    __syncthreads();

    float* F = f2 + ((size_t)b*32 + s) * 256;
    for (int i = tid; i < 256; i += 128) F[i] = CM[i];
}

// --------------------------- SA stage 3 + concat ----------------------------
// block = b. X = [xyz2 | f2] (32 x 259) -> 256 -> 256 -> 512 -> max => f3,
// then X1[b] = [f3 | one_hot | 0]  (stride 516, zero pad col)
__global__ void sa3_kernel(const float* __restrict__ xyz2, const float* __restrict__ f2,
                           const float* __restrict__ onehot,
                           const float* __restrict__ Wp,
                           const float* __restrict__ S, const float* __restrict__ T,
                           float* __restrict__ X1) {
    extern __shared__ float smem[];
    float* X  = smem;                  // 32 x 265 (0-2 xyz, 3-258 feats, 259-263 zero)
    float* B1 = X  + 32 * 265;         // 32 x 257
    float* B2 = B1 + 32 * 257;         // 32 x 257
    float* CM = B2 + 32 * 257;         // 512

    const int tid = threadIdx.x, lane = tid & 31, wave = tid >> 5;
    const int b = blockIdx.x;
    const float* P2  = xyz2 + (size_t)b * 32 * 3;
    const float* F2b = f2   + (size_t)b * 32 * 256;

    for (int i = tid; i < 512; i += 128) CM[i] = 0.f;
    for (int e = tid; e < 32 * 264; e += 128) {
        int row = e / 264, col = e - row * 264;
        float v;
        if (col < 3)        v = P2[row*3 + col];
        else if (col < 259) v = F2b[(size_t)row*256 + (col - 3)];
        else                v = 0.f;
        X[(size_t)row*265 + col] = v;
    }
    __syncthreads();

    mlp_tiles<false, 65>(X, 265, 32, Wp + 79360,  256, S+768,  T+768,  B1, 257, wave, lane, 4);
    __syncthreads();
    mlp_tiles<false, 64>(B1, 257, 32, Wp + 146944, 256, S+1024, T+1024, B2, 257, wave, lane, 4);
    __syncthreads();
    mlp_tiles<true,  64>(B2, 257, 32, Wp + 212480, 512, S+1280, T+1280, CM, 0,   wave, lane, 4);
    __syncthreads();

    float* O = X1 + (size_t)b * 516;
    for (int i = tid; i < 512; i += 128) O[i] = CM[i];
    if (tid < 4) O[512 + tid] = (tid < 3) ? onehot[b*3 + tid] : 0.f;
}

// --------------------------- batched FC GEMM (M=128) ------------------------
// X: lda padded so 4*KCH <= lda reads stay in zero-padded region.
template <int KCH, bool RELU>
__global__ void fc_gemm_kernel(const float* __restrict__ X, int lda, int M,
                               const float* __restrict__ Wp, int Cout,
                               const float* __restrict__ S, const float* __restrict__ T,
                               float* __restrict__ Y, int ldy) {
    constexpr int KPAIRS = KCH >> 1;
    constexpr int KPT    = (KCH + 1) >> 1;
    const int lane = threadIdx.x & 31, wave = threadIdx.x >> 5;
    const int mt_n = M >> 4;
    const int nt_n = (Cout + 15) >> 4;
    const int tile = blockIdx.x * 4 + wave;
    if (tile >= mt_n * nt_n) return;
    const int m0 = (tile % mt_n) << 4;
    const int nt = tile / mt_n;
    const int mrow  = lane & 15;
    const int khalf = (lane >> 4) << 1;
    const float* Arow = X + (size_t)(m0 + mrow) * lda + khalf;
    const v4f*   Bq   = (const v4f*)Wp + (size_t)nt * KPT * 32 + lane;

    v8f acc0 = {0.f,0.f,0.f,0.f,0.f,0.f,0.f,0.f};
    v8f acc1 = {0.f,0.f,0.f,0.f,0.f,0.f,0.f,0.f};
#pragma unroll 8
    for (int kp = 0; kp < KPAIRS; ++kp) {
        v4f bb = Bq[(size_t)kp * 32];
        const float* Ac = Arow + kp * 8;
        v2f a0; a0.x = Ac[0]; a0.y = Ac[1];
        v2f a1; a1.x = Ac[4]; a1.y = Ac[5];
        v2f b0; b0.x = bb.x;  b0.y = bb.y;
        v2f b1; b1.x = bb.z;  b1.y = bb.w;
        acc0 = __builtin_amdgcn_wmma_f32_16x16x4_f32(
                   false, a0, false, b0, (short)0, acc0, false, false);
        acc1 = __builtin_amdgcn_wmma_f32_16x16x4_f32(
                   false, a1, false, b1, (short)0, acc1, false, false);
    }
    if constexpr (KCH & 1) {
        v4f bb = Bq[(size_t)KPAIRS * 32];
        const float* Ac = Arow + KPAIRS * 8;
        v2f a; a.x = Ac[0]; a.y = Ac[1];
        v2f b0; b0.x = bb.x; b0.y = bb.y;
        acc0 = __builtin_amdgcn_wmma_f32_16x16x4_f32(
                   false, a, false, b0, (short)0, acc0, false, false);
    }
    v8f acc = acc0 + acc1;

    const int  n   = (nt << 4) + mrow;
    const bool ncv = (n < Cout);
    const float sc = ncv ? S[n] : 0.f;
    const float tc = ncv ? T[n] : 0.f;
    const int   mh = (lane >> 4) << 3;
#pragma unroll
    for (int r = 0; r < 8; ++r) {
        float v = acc[r] * sc + tc;
        if (RELU) v = fmaxf(v, 0.f);
        if (ncv) Y[(size_t)(m0 + mh + r) * ldy + n] = v;
    }
}

// ---------------------------------------------------------------------------
extern "C" void kernel_launch(void* const* d_in, const int* in_sizes, int n_in,
                              void* d_out, int out_size, void* d_ws, size_t ws_size,
                              hipStream_t stream) {
    const float* pts    = (const float*)d_in[0];
    const float* onehot = (const float*)d_in[1];

    // params flattened in jax pytree order: top dict keys sorted
    // (fc, sa1, sa2, sa3); layer dicts sorted (W, b, beta, gamma, mean, var)
    WPtrs  wp;
    BNPtrs bp;
    const int gbase[4] = {20, 38, 56, 2};   // sa1, sa2, sa3, fc start indices
    for (int g = 0; g < 4; ++g) {
        for (int l = 0; l < 3; ++l) {
            int ly = g * 3 + l;
            int bi = gbase[g] + 6 * l;
            wp.W[ly]     = (const float*)d_in[bi + 0];
            bp.b[ly]     = (const float*)d_in[bi + 1];
            bp.beta[ly]  = (const float*)d_in[bi + 2];
            bp.gamma[ly] = (const float*)d_in[bi + 3];
            bp.mean[ly]  = (const float*)d_in[bi + 4];
            bp.var[ly]   = (const float*)d_in[bi + 5];
        }
    }

    float* ws   = (float*)d_ws;
    float* S    = ws + OFF_S;
    float* T    = ws + OFF_T;
    float* xyz1 = ws + OFF_XYZ1;
    float* xyz2 = ws + OFF_XYZ2;
    float* f1   = ws + OFF_F1;
    float* f2   = ws + OFF_F2;
    float* X1   = ws + OFF_X1;
    float* H1   = ws + OFF_H1;
    float* H2   = ws + OFF_H2;
    float* Wpk  = ws + OFF_WP;

    fuse_bn_kernel<<<12, 512, 0, stream>>>(bp, S, T);
    pack_w_kernel<<<12, 256, 0, stream>>>(wp, Wpk);

    fps_kernel<NPOINTS, 128, 256><<<BATCH, 256, 0, stream>>>(pts, xyz1);

    size_t sh1 = (size_t)(64*9 + 2*64*65 + 128) * sizeof(float) + 64 * sizeof(int);
    sa1_kernel<<<BATCH * 128, 128, sh1, stream>>>(pts, xyz1, Wpk, S, T, f1);

    fps_kernel<128, 32, 128><<<BATCH, 128, 0, stream>>>(xyz1, xyz2);

    size_t sh2 = (size_t)(64*137 + 2*64*129 + 256) * sizeof(float) + 64 * sizeof(int);
    sa2_kernel<<<BATCH * 32, 128, sh2, stream>>>(xyz1, f1, xyz2, Wpk, S, T, f2);

    size_t sh3 = (size_t)(32*265 + 2*32*257 + 512) * sizeof(float);
    sa3_kernel<<<BATCH, 128, sh3, stream>>>(xyz2, f2, onehot, Wpk, S, T, X1);

    // fc head: 128x515 @ 515x512 -> relu -> @512x256 -> relu -> @256x59
    fc_gemm_kernel<129, true ><<<64, 128, 0, stream>>>(X1, 516, BATCH, Wpk + h_POFF[9],  512,
                                                       S+1792, T+1792, H1, 512);
    fc_gemm_kernel<128, true ><<<32, 128, 0, stream>>>(H1, 512, BATCH, Wpk + h_POFF[10], 256,
                                                       S+2304, T+2304, H2, 256);
    fc_gemm_kernel< 64, false><<< 8, 128, 0, stream>>>(H2, 256, BATCH, Wpk + h_POFF[11],  59,
                                                       S+2560, T+2560, (float*)d_out, 59);
}